// EdgeSoftmax_74182675137182
// MI455X (gfx1250) — compile-verified
//
#include <hip/hip_runtime.h>

#define THREADS 256

typedef float    v4f __attribute__((ext_vector_type(4)));
typedef unsigned v4u __attribute__((ext_vector_type(4)));

// ---- monotonic float <-> uint32 key mapping (order-preserving, incl. negatives) ----
__device__ __forceinline__ unsigned f2key(float f) {
    unsigned b = __float_as_uint(f);
    return b ^ ((unsigned)((int)b >> 31) | 0x80000000u);
}
__device__ __forceinline__ float key2f(unsigned u) {
    unsigned mask = (unsigned)((int)(~u) >> 31) | 0x80000000u;
    return __uint_as_float(u ^ mask);
}

// Zero the per-node max-key table (ws) and the normalizer output region.
// key 0 is below the key of every real float (acts as -inf), norm starts at 0.
__global__ __launch_bounds__(THREADS)
void EdgeSoftmax_init_kernel(unsigned* __restrict__ keys,
                             float* __restrict__ norm, int nh) {
    int i = blockIdx.x * blockDim.x + threadIdx.x;
    if (i < nh) {
        keys[i] = 0u;
        norm[i] = 0.0f;
    }
}

// ---------------- H == 8 fast path: one thread per edge, b128 vector ops ----------------

__global__ __launch_bounds__(THREADS)
void EdgeSoftmax_pass1_h8(const v4f* __restrict__ logits4,
                          const int* __restrict__ dst,
                          unsigned* __restrict__ keys, int E) {
    int e = blockIdx.x * blockDim.x + threadIdx.x;
    if (e >= E) return;
    // 32 contiguous bytes per edge -> two global_load_b128, wave-coalesced
    v4f a = logits4[2 * e];
    v4f b = logits4[2 * e + 1];
    int d = dst[e];
    unsigned* __restrict__ k = keys + (size_t)d * 8;
    // 8 fire-and-forget global_atomic_max_u32 into the L2-resident 3.2MB table
    atomicMax(&k[0], f2key(a.x));
    atomicMax(&k[1], f2key(a.y));
    atomicMax(&k[2], f2key(a.z));
    atomicMax(&k[3], f2key(a.w));
    atomicMax(&k[4], f2key(b.x));
    atomicMax(&k[5], f2key(b.y));
    atomicMax(&k[6], f2key(b.z));
    atomicMax(&k[7], f2key(b.w));
}

__global__ __launch_bounds__(THREADS)
void EdgeSoftmax_pass2_h8(const v4f* __restrict__ logits4,
                          const int* __restrict__ dst,
                          const unsigned* __restrict__ keys,
                          v4f* __restrict__ scores4,
                          float* __restrict__ norm, int E) {
    int e = blockIdx.x * blockDim.x + threadIdx.x;
    if (e >= E) return;
    // logits re-read: expected to hit the 192MB L2 (brought in by pass 1)
    v4f a = logits4[2 * e];
    v4f b = logits4[2 * e + 1];
    int d = dst[e];
    const v4u* __restrict__ k4 = (const v4u*)(keys + (size_t)d * 8);
    v4u k0 = k4[0];
    v4u k1 = k4[1];

    v4f s0, s1;
    s0.x = __expf(a.x - key2f(k0.x));
    s0.y = __expf(a.y - key2f(k0.y));
    s0.z = __expf(a.z - key2f(k0.z));
    s0.w = __expf(a.w - key2f(k0.w));
    s1.x = __expf(b.x - key2f(k1.x));
    s1.y = __expf(b.y - key2f(k1.y));
    s1.z = __expf(b.z - key2f(k1.z));
    s1.w = __expf(b.w - key2f(k1.w));

    // write-once 102MB stream: non-temporal so it doesn't evict logits/dst from L2
    __builtin_nontemporal_store(s0, scores4 + 2 * e);
    __builtin_nontemporal_store(s1, scores4 + 2 * e + 1);

    // native fp32 agent-scope atomics -> global_atomic_add_f32 (no CAS loop)
    float* __restrict__ n = norm + (size_t)d * 8;
    __hip_atomic_fetch_add(&n[0], s0.x, __ATOMIC_RELAXED, __HIP_MEMORY_SCOPE_AGENT);
    __hip_atomic_fetch_add(&n[1], s0.y, __ATOMIC_RELAXED, __HIP_MEMORY_SCOPE_AGENT);
    __hip_atomic_fetch_add(&n[2], s0.z, __ATOMIC_RELAXED, __HIP_MEMORY_SCOPE_AGENT);
    __hip_atomic_fetch_add(&n[3], s0.w, __ATOMIC_RELAXED, __HIP_MEMORY_SCOPE_AGENT);
    __hip_atomic_fetch_add(&n[4], s1.x, __ATOMIC_RELAXED, __HIP_MEMORY_SCOPE_AGENT);
    __hip_atomic_fetch_add(&n[5], s1.y, __ATOMIC_RELAXED, __HIP_MEMORY_SCOPE_AGENT);
    __hip_atomic_fetch_add(&n[6], s1.z, __ATOMIC_RELAXED, __HIP_MEMORY_SCOPE_AGENT);
    __hip_atomic_fetch_add(&n[7], s1.w, __ATOMIC_RELAXED, __HIP_MEMORY_SCOPE_AGENT);
}

// ---------------- generic-H fallback: one thread per (edge, head) ----------------

__global__ __launch_bounds__(THREADS)
void EdgeSoftmax_pass1_gen(const float* __restrict__ logits,
                           const int* __restrict__ dst,
                           unsigned* __restrict__ keys, int EH, int H) {
    int i = blockIdx.x * blockDim.x + threadIdx.x;
    if (i >= EH) return;
    int e = i / H;
    int h = i - e * H;
    atomicMax(&keys[(size_t)dst[e] * H + h], f2key(logits[i]));
}

__global__ __launch_bounds__(THREADS)
void EdgeSoftmax_pass2_gen(const float* __restrict__ logits,
                           const int* __restrict__ dst,
                           const unsigned* __restrict__ keys,
                           float* __restrict__ scores,
                           float* __restrict__ norm, int EH, int H) {
    int i = blockIdx.x * blockDim.x + threadIdx.x;
    if (i >= EH) return;
    int e = i / H;
    int h = i - e * H;
    size_t nidx = (size_t)dst[e] * H + h;
    float s = __expf(logits[i] - key2f(keys[nidx]));
    __builtin_nontemporal_store(s, &scores[i]);
    __hip_atomic_fetch_add(&norm[nidx], s, __ATOMIC_RELAXED, __HIP_MEMORY_SCOPE_AGENT);
}

extern "C" void kernel_launch(void* const* d_in, const int* in_sizes, int n_in,
                              void* d_out, int out_size, void* d_ws, size_t ws_size,
                              hipStream_t stream) {
    const float* logits = (const float*)d_in[0];  // [E, H, 1] f32
    const int*   dst    = (const int*)d_in[1];    // [E] i32
    (void)n_in; (void)ws_size;

    const int EH = in_sizes[0];      // E * H
    const int E  = in_sizes[1];
    const int H  = EH / E;
    const int NH = out_size - EH;    // N * H (normalizer element count)

    float* scores = (float*)d_out;        // [E, H]
    float* norm   = (float*)d_out + EH;   // [N, H]
    unsigned* keys = (unsigned*)d_ws;     // [N, H] monotonic max keys (3.2 MB)

    const int initBlocks = (NH + THREADS - 1) / THREADS;
    EdgeSoftmax_init_kernel<<<initBlocks, THREADS, 0, stream>>>(keys, norm, NH);

    if (H == 8) {
        const int blocks = (E + THREADS - 1) / THREADS;
        EdgeSoftmax_pass1_h8<<<blocks, THREADS, 0, stream>>>(
            (const v4f*)logits, dst, keys, E);
        EdgeSoftmax_pass2_h8<<<blocks, THREADS, 0, stream>>>(
            (const v4f*)logits, dst, keys, (v4f*)scores, norm, E);
    } else {
        const int blocks = (EH + THREADS - 1) / THREADS;
        EdgeSoftmax_pass1_gen<<<blocks, THREADS, 0, stream>>>(
            logits, dst, keys, EH, H);
        EdgeSoftmax_pass2_gen<<<blocks, THREADS, 0, stream>>>(
            logits, dst, keys, scores, norm, EH, H);
    }
}